// SplineNetwork_35691178230289
// MI455X (gfx1250) — compile-verified
//
#include <hip/hip_runtime.h>
#include <stdint.h>

// SplineNetwork on MI455X (gfx1250).
// Inputs : x (16384,2) f32, weights (16384,1) f32, control_points (16384,2) f32
// Output : concat( out (16384,1) f32 , x (16384,2) f32 )  -> 49152 f32
//
// Algorithm: control points are a regular 128x128 grid over [-1,1]^2, so the
// 9 nearest neighbors of any query are contained in the 4x4 grid window around
// the containing cell. Per point: 16 candidate distances (separable), branchless
// rank<9 selection (tie-break by lower flat index, matching jax top_k), Keys
// cubic kernel, weighted sum. Weight table (64KB) and the 128-entry linspace
// table are staged into LDS with CDNA5 async global->LDS DMA (ASYNCcnt).

#define NGRID 128
#define BATCH 16384
#define KNN   9
#define TPB   256

__device__ __forceinline__ uint32_t lds_off_u32(const void* p) {
    // flat address of a __shared__ object: low 32 bits are the LDS byte offset
    return (uint32_t)(uintptr_t)p;
}

__global__ __launch_bounds__(TPB) void spline_knn_kernel(
    const float* __restrict__ x,    // (B,2)
    const float* __restrict__ w,    // (M,1)
    const float* __restrict__ cp,   // (M,2)
    float* __restrict__ out)        // [0,B): out ; [B, 3B): copy of x
{
    __shared__ __attribute__((aligned(16))) float s_w[NGRID * NGRID]; // 64 KB
    __shared__ __attribute__((aligned(16))) float s_t[NGRID];         // 512 B

    const int lt  = threadIdx.x;
    const int tid = blockIdx.x * TPB + lt;

    // ---- async stage: full weight table, 16 x b128 per thread ----
#pragma unroll
    for (int i = 0; i < 16; ++i) {
        const int idx4 = i * TPB + lt;                 // float4 index
        const uint32_t loff = lds_off_u32(&s_w[idx4 * 4]);
        const float* gp = w + (size_t)idx4 * 4;
        asm volatile("global_load_async_to_lds_b128 %0, %1, off"
                     :: "v"(loff), "v"(gp) : "memory");
    }
    // ---- async stage: 1-D linspace table t[j] = control_points[j].y ----
    if (lt < NGRID) {
        const uint32_t loff = lds_off_u32(&s_t[lt]);
        const float* gp = cp + 2 * lt + 1;
        asm volatile("global_load_async_to_lds_b32 %0, %1, off"
                     :: "v"(loff), "v"(gp) : "memory");
    }
    asm volatile("s_wait_asynccnt 0" ::: "memory");
    __syncthreads();

    // ---- per-point compute ----
    const float qx = x[2 * tid + 0];
    const float qy = x[2 * tid + 1];

    // data-dependent bandwidth h = ||x[0] - x[1]|| (uniform across grid)
    const float hx = x[0] - x[2];
    const float hy = x[1] - x[3];
    const float inv_h = 1.0f / sqrtf(hx * hx + hy * hy);

    // 4x4 candidate window around containing cell (one-cell slack each side)
    const float scale = (NGRID - 1) * 0.5f;            // 63.5
    int ic = (int)floorf((qx + 1.0f) * scale);
    int jc = (int)floorf((qy + 1.0f) * scale);
    int i0 = min(max(ic - 1, 0), NGRID - 4);
    int j0 = min(max(jc - 1, 0), NGRID - 4);

    float ti[4], tj[4];
#pragma unroll
    for (int k = 0; k < 4; ++k) {
        ti[k] = s_t[i0 + k];
        tj[k] = s_t[j0 + k];
    }

    // selection metric mimics reference: D = (|q|^2 + |c|^2) - 2 q.c
    const float q2 = qx * qx + qy * qy;
    float D[16];     // ranking metric
    float dt[16];    // true squared distance (for s)
    int   id[16];    // flat control-point index
#pragma unroll
    for (int ii = 0; ii < 4; ++ii) {
#pragma unroll
        for (int jj = 0; jj < 4; ++jj) {
            const int c = ii * 4 + jj;
            const float cx = ti[ii], cy = tj[jj];
            const float c2  = cx * cx + cy * cy;
            const float dot = qx * cx + qy * cy;
            D[c]  = (q2 + c2) - 2.0f * dot;
            const float dx = qx - cx, dy = qy - cy;
            dt[c] = dx * dx + dy * dy;
            id[c] = (i0 + ii) * NGRID + (j0 + jj);
        }
    }

    // branchless rank-based 9-NN select; tie-break: lower index wins (top_k)
    float acc = 0.0f;
#pragma unroll
    for (int c = 0; c < 16; ++c) {
        int rank = 0;
#pragma unroll
        for (int o = 0; o < 16; ++o) {
            const bool less = (D[o] < D[c]) || ((D[o] == D[c]) && (id[o] < id[c]));
            rank += less ? 1 : 0;
        }
        if (rank < KNN) {
            const float s  = sqrtf(dt[c]) * inv_h;
            const float s2 = s * s;
            const float s3 = s2 * s;
            const float p1 =  1.5f * s3 - 2.5f * s2 + 1.0f;
            const float p2 = -0.5f * s3 + 2.5f * s2 - 4.0f * s + 2.0f;
            const float cv = (s < 1.0f) ? p1 : (((s > 1.0f) && (s < 2.0f)) ? p2 : 0.0f);
            acc += s_w[id[c]] * cv;
        }
    }

    out[tid] = acc;
    out[BATCH + 2 * tid + 0] = qx;
    out[BATCH + 2 * tid + 1] = qy;
}

extern "C" void kernel_launch(void* const* d_in, const int* in_sizes, int n_in,
                              void* d_out, int out_size, void* d_ws, size_t ws_size,
                              hipStream_t stream) {
    const float* x  = (const float*)d_in[0];
    const float* w  = (const float*)d_in[1];
    const float* cp = (const float*)d_in[2];
    float* out = (float*)d_out;
    (void)in_sizes; (void)n_in; (void)out_size; (void)d_ws; (void)ws_size;

    spline_knn_kernel<<<BATCH / TPB, TPB, 0, stream>>>(x, w, cp, out);
}